// SparseECA_25683904430831
// MI455X (gfx1250) — compile-verified
//
#include <hip/hip_runtime.h>

// SparseECA for MI455X (gfx1250, wave32).
// Pass 1: segment-sum as one-hot matmul via V_WMMA_F32_16X16X4_F32.
// Pass 2: counts (binary search on sorted batch_idx) + mean + conv1d(k=3,SAME) + sigmoid.
// Pass 3: float4 streaming apply: out = feat * scale[batch_idx].
// Roofline: ~1.15 GB HBM traffic -> ~49 us at 23.3 TB/s. Memory bound; fp32 throughout.

typedef __attribute__((ext_vector_type(2))) float v2f;
typedef __attribute__((ext_vector_type(8))) float v8f;

#define CCH 96   // channels (reference constant)
#define NB  8    // batches  (reference constant)

__global__ void k_zero(float* __restrict__ p, int n) {
  int i = blockIdx.x * blockDim.x + threadIdx.x;
  if (i < n) p[i] = 0.0f;
}

// One block = 6 waves; wave w owns channel tile [16w, 16w+16).
// Per iteration a wave consumes 4 rows x 16 channels:
//   A[16x4] = one-hot(batch_idx[row..row+3])  (M = batch, K = row-in-quad)
//   B[4x16] = features tile
//   D[16x16] += A x B   -> per-batch channel sums in the accumulator.
__global__ __launch_bounds__(192) void k_segsum_wmma(
    const float* __restrict__ feat, const int* __restrict__ bidx,
    float* __restrict__ sums, int nQuads) {
  const int lane = threadIdx.x & 31;
  const int wave = threadIdx.x >> 5;      // 0..5
  const int cb   = wave << 4;             // channel base of this wave's tile
  const int m    = lane & 15;             // M (batch row) for A, N (channel) for B
  const int k0   = (lane >> 4) << 1;      // lanes 0-15 -> K{0,1}; lanes 16-31 -> K{2,3}

  v8f acc = {0.f, 0.f, 0.f, 0.f, 0.f, 0.f, 0.f, 0.f};

  for (int q = blockIdx.x; q < nQuads; q += gridDim.x) {
    const int row = q << 2;
    const int4 bi = *(const int4*)(bidx + row);        // broadcast, L1 hit
    const int i0 = (k0 == 0) ? bi.x : bi.z;
    const int i1 = (k0 == 0) ? bi.y : bi.w;

    v2f a, b;
    a.x = (i0 == m) ? 1.0f : 0.0f;                     // A: one-hot selector
    a.y = (i1 == m) ? 1.0f : 0.0f;
    const float* fr = feat + (size_t)(row + k0) * CCH + cb + m;
    b.x = fr[0];                                       // B[k0  ][m]
    b.y = fr[CCH];                                     // B[k0+1][m]

    acc = __builtin_amdgcn_wmma_f32_16x16x4_f32(
        false, a, false, b, (short)0, acc, false, false);
  }

  // D layout: VGPR r, lanes 0-15 -> M=r (batches 0..7), N=lane. Batches 8..15 are zero.
  if (lane < 16) {
#pragma unroll
    for (int r = 0; r < NB; ++r) {
      __hip_atomic_fetch_add(&sums[r * CCH + cb + lane], acc[r],
                             __ATOMIC_RELAXED, __HIP_MEMORY_SCOPE_AGENT);
    }
  }
}

// Single block: tail rows (N%4), counts via binary search, mean, conv1d+sigmoid.
__global__ __launch_bounds__(128) void k_eca(
    const float* __restrict__ feat, const int* __restrict__ bidx,
    const float* __restrict__ convw, float* __restrict__ sums,
    float* __restrict__ scale, int N, int nTail) {
  __shared__ float s_mean[NB * CCH];
  __shared__ int   s_lb[NB + 1];
  const int tid = threadIdx.x;

  if (tid < CCH) {                       // fold in tail rows (nTail <= 3)
    for (int r = N - nTail; r < N; ++r)
      sums[bidx[r] * CCH + tid] += feat[(size_t)r * CCH + tid];
  }
  if (tid <= NB) {                       // lower_bound(batch_idx, tid)
    int lo = 0, hi = N;
    while (lo < hi) {
      int mid = (lo + hi) >> 1;
      if (bidx[mid] < tid) lo = mid + 1; else hi = mid;
    }
    s_lb[tid] = lo;
  }
  __syncthreads();

  for (int i = tid; i < NB * CCH; i += blockDim.x) {
    int b = i / CCH;
    float cnt = (float)(s_lb[b + 1] - s_lb[b]);
    s_mean[i] = sums[i] / fmaxf(cnt, 1.0f);
  }
  __syncthreads();

  const float w0 = convw[0], w1 = convw[1], w2 = convw[2];
  for (int i = tid; i < NB * CCH; i += blockDim.x) {
    int b = i / CCH, c = i - b * CCH;
    float xl = (c > 0)       ? s_mean[i - 1] : 0.0f;   // SAME zero padding
    float xr = (c < CCH - 1) ? s_mean[i + 1] : 0.0f;
    float y  = w0 * xl + w1 * s_mean[i] + w2 * xr;     // lax conv = cross-correlation
    scale[i] = 1.0f / (1.0f + __expf(-y));
  }
}

// float4 stream: 24 float4 per row (C=96). scale table (768 f) stays cache-hot.
__global__ __launch_bounds__(256) void k_apply(
    const float* __restrict__ feat, const int* __restrict__ bidx,
    const float* __restrict__ scale, float* __restrict__ out, long long total4) {
  long long j = (long long)blockIdx.x * blockDim.x + threadIdx.x;
  if (j >= total4) return;
  int row = (int)(j / 24);
  int c4  = (int)(j - (long long)row * 24);
  int b   = bidx[row];
  const float4 f = ((const float4*)feat)[j];
  const float4 s = ((const float4*)(scale + b * CCH))[c4];
  float4 o;
  o.x = f.x * s.x; o.y = f.y * s.y; o.z = f.z * s.z; o.w = f.w * s.w;
  ((float4*)out)[j] = o;
}

extern "C" void kernel_launch(void* const* d_in, const int* in_sizes, int n_in,
                              void* d_out, int out_size, void* d_ws, size_t ws_size,
                              hipStream_t stream) {
  const float* feat  = (const float*)d_in[0];   // [N, 96] fp32
  const int*   bidx  = (const int*)d_in[1];     // [N] sorted int32
  const float* convw = (const float*)d_in[3];   // [1,1,3] fp32
  float* out = (float*)d_out;

  const int N = in_sizes[1];

  float* sums  = (float*)d_ws;                  // 768 floats
  float* scale = sums + NB * CCH;               // 768 floats

  k_zero<<<(NB * CCH + 255) / 256, 256, 0, stream>>>(sums, NB * CCH);

  const int nQuads = N / 4;
  const int nTail  = N & 3;
  k_segsum_wmma<<<768, 192, 0, stream>>>(feat, bidx, sums, nQuads);

  k_eca<<<1, 128, 0, stream>>>(feat, bidx, convw, sums, scale, N, nTail);

  const long long total4 = (long long)N * (CCH / 4);
  const int blocks = (int)((total4 + 255) / 256);
  k_apply<<<blocks, 256, 0, stream>>>(feat, bidx, scale, out, total4);
}